// SimpleALIFRNNTbptt_52750788329833
// MI455X (gfx1250) — compile-verified
//
#include <hip/hip_runtime.h>

typedef __attribute__((ext_vector_type(2))) float v2f;
typedef __attribute__((ext_vector_type(8))) float v8f;

#define T_STEPS 250
#define BATCH   128
#define INDIM   700
#define HDIM    512
#define ODIM    20
#define KC      1212           // INDIM + HDIM
#define KP      1216           // padded K (multiple of 4), pad cols are zero
#define NWG     32
#define NTHR    256
#define GTHREADS (NWG*NTHR)
#define B0C     0.01f
#define BETAC   1.8f
#define LRC     0.01f
#define SLOPEC  10.0f

// ---- workspace layout (offsets in floats) ----
#define OFF_WHT   0ull          // W_h^T  [HDIM][KP]   (K-contiguous B operand)
#define OFF_ACAT  622592ull     // cat(x_t, z) [BATCH][KP]
#define OFF_ZNEW  778240ull     // z_{t}   [BATCH][HDIM]
#define OFF_ZOLD  843776ull     // z_{t-1} checkpoint (update steps only)
#define OFF_U0    909312ull
#define OFF_U1    974848ull
#define OFF_A0    1040384ull
#define OFF_A1    1105920ull
#define OFF_IWS   1171456ull    // I checkpoint (update steps only)
#define OFF_GV    1236992ull
#define OFF_OU0   1302528ull
#define OFF_OU1   1305088ull
#define OFF_R     1307648ull    // r = z_new @ W_o (pre-decay readout)
#define OFF_GOU   1310208ull
#define OFF_WO    1312768ull
#define OFF_TAUM  1323008ull
#define OFF_TAUA  1323520ull
#define OFF_TAUO  1324032ull
#define OFF_AL    1324064ull    // cached alpha (pre-update)
#define OFF_RH    1324576ull
#define OFF_ALO   1325088ull
#define OFF_NSPK  1325120ull
#define OFF_BAR   1325124ull    // 2 x unsigned barrier state

__device__ inline void grid_sync(unsigned* bar) {
  __threadfence();
  __syncthreads();
  if (threadIdx.x == 0) {
    unsigned* arrive = bar;
    unsigned* gen    = bar + 1;
    unsigned g = __hip_atomic_load(gen, __ATOMIC_RELAXED, __HIP_MEMORY_SCOPE_AGENT);
    unsigned a = __hip_atomic_fetch_add(arrive, 1u, __ATOMIC_ACQ_REL, __HIP_MEMORY_SCOPE_AGENT);
    if (a == NWG - 1) {
      __hip_atomic_store(arrive, 0u, __ATOMIC_RELAXED, __HIP_MEMORY_SCOPE_AGENT);
      __hip_atomic_fetch_add(gen, 1u, __ATOMIC_ACQ_REL, __HIP_MEMORY_SCOPE_AGENT);
    } else {
      while (__hip_atomic_load(gen, __ATOMIC_ACQUIRE, __HIP_MEMORY_SCOPE_AGENT) == g) {
        __builtin_amdgcn_s_sleep(2);
      }
    }
  }
  __syncthreads();
  __threadfence();
}

__global__ void alif_init(const float* __restrict__ Wh, const float* __restrict__ tm,
                          const float* __restrict__ ta, const float* __restrict__ Wo_in,
                          const float* __restrict__ tmo, float* __restrict__ ws) {
  const size_t i0 = (size_t)blockIdx.x * blockDim.x + threadIdx.x;
  const size_t stride = (size_t)gridDim.x * blockDim.x;
  float* WhT  = ws + OFF_WHT;
  float* Acat = ws + OFF_ACAT;
  float* Znew = ws + OFF_ZNEW;
  float* u0   = ws + OFF_U0;
  float* a0   = ws + OFF_A0;
  float* ou0  = ws + OFF_OU0;
  float* Wo   = ws + OFF_WO;
  float* tauM = ws + OFF_TAUM;
  float* tauA = ws + OFF_TAUA;
  float* tauO = ws + OFF_TAUO;
  for (size_t i = i0; i < (size_t)HDIM * KP; i += stride) {
    size_t h = i / KP, k = i - h * KP;
    WhT[i] = (k < KC) ? Wh[k * HDIM + h] : 0.f;   // transpose + zero K-pad
  }
  for (size_t i = i0; i < (size_t)BATCH * HDIM; i += stride) {
    u0[i] = 0.f; a0[i] = 0.f; Znew[i] = 0.f;
  }
  for (size_t i = i0; i < (size_t)BATCH * ODIM; i += stride) ou0[i] = 0.f;
  for (size_t i = i0; i < (size_t)HDIM * ODIM; i += stride) Wo[i] = Wo_in[i];
  for (size_t i = i0; i < HDIM; i += stride) { tauM[i] = tm[i]; tauA[i] = ta[i]; }
  for (size_t i = i0; i < ODIM; i += stride) tauO[i] = tmo[i];
  for (size_t i = i0; i < (size_t)BATCH * (KP - KC); i += stride) {
    size_t b = i / (KP - KC), j = i - b * (KP - KC);
    Acat[b * KP + KC + j] = 0.f;                   // zero A K-pad (never rewritten)
  }
  if (i0 == 0) {
    ws[OFF_NSPK] = 0.f;
    unsigned* bar = (unsigned*)(ws + OFF_BAR);
    bar[0] = 0u; bar[1] = 0u;
  }
}

__global__ __launch_bounds__(NTHR, 1) void alif_run(
    const float* __restrict__ x, const int* __restrict__ y,
    float* __restrict__ out, float* ws) {
  float* WhT   = ws + OFF_WHT;
  float* Acat  = ws + OFF_ACAT;
  float* Znew  = ws + OFF_ZNEW;
  float* Zold  = ws + OFF_ZOLD;
  float* ubuf[2]  = { ws + OFF_U0,  ws + OFF_U1 };
  float* abuf[2]  = { ws + OFF_A0,  ws + OFF_A1 };
  float* oubuf[2] = { ws + OFF_OU0, ws + OFF_OU1 };
  float* Iws   = ws + OFF_IWS;
  float* gv    = ws + OFF_GV;
  float* rbuf  = ws + OFF_R;
  float* gou   = ws + OFF_GOU;
  float* Wo    = ws + OFF_WO;
  float* tauM  = ws + OFF_TAUM;
  float* tauA  = ws + OFF_TAUA;
  float* tauO  = ws + OFF_TAUO;
  float* alC   = ws + OFF_AL;
  float* rhC   = ws + OFF_RH;
  float* aoC   = ws + OFF_ALO;
  float* nspk  = ws + OFF_NSPK;
  unsigned* bar = (unsigned*)(ws + OFF_BAR);

  const int tid  = threadIdx.x;
  const int gtid = blockIdx.x * NTHR + tid;
  const int wid  = tid >> 5;          // wave in block: 0..7  -> M tile
  const int lane = tid & 31;
  const int mt   = wid;               // 8 M-tiles (BATCH=128)
  const int nt   = blockIdx.x;        // 32 N-tiles (HDIM=512)
  const int lr   = lane & 15;
  const int lh   = lane >> 4;

  __shared__ float redS[NTHR];
  __shared__ float lossS[BATCH];
  __shared__ float bcastS;

  float lv = 0.f, tot = 0.f;

  // build A_cat for t=0: x[0] | z=0
  for (int i = gtid; i < BATCH * INDIM; i += GTHREADS) {
    int b = i / INDIM, j = i - b * INDIM;
    Acat[(size_t)b * KP + j] = x[i];
  }
  for (int i = gtid; i < BATCH * HDIM; i += GTHREADS) {
    int b = i >> 9, h = i & (HDIM - 1);
    Acat[(size_t)b * KP + INDIM + h] = Znew[i];
  }
  grid_sync(bar);

  const float* Arow = Acat + (size_t)(mt * 16 + lr) * KP + 2 * lh;  // A 16x4 f32 frag rows
  const float* Brow = WhT  + (size_t)(nt * 16 + lr) * KP + 2 * lh;  // B 4x16 f32 frag cols

  for (int t = 0; t < T_STEPS; ++t) {
    const int po = t & 1, pn = po ^ 1;
    const bool periodic = ((t % 50) == 0) || (t == T_STEPS - 1);
    float* u_old = ubuf[po];  float* u_new = ubuf[pn];
    float* a_old = abuf[po];  float* a_new = abuf[pn];
    float* ou_old = oubuf[po]; float* ou_new = oubuf[pn];

    // ---- I = cat(x_t, z) @ W_h : one 16x16 tile per wave, f32 WMMA, K=1216 ----
    v8f acc = {0.f, 0.f, 0.f, 0.f, 0.f, 0.f, 0.f, 0.f};
    #pragma unroll 8
    for (int k0 = 0; k0 < KP; k0 += 4) {
      v2f av = *(const v2f*)(Arow + k0);
      v2f bv = *(const v2f*)(Brow + k0);
      acc = __builtin_amdgcn_wmma_f32_16x16x4_f32(false, av, false, bv,
                                                  (short)0, acc, false, false);
    }

    // ---- elementwise ALIF update directly on accumulator tile ----
    {
      const int hcol = nt * 16 + lr;
      const float tm = tauM[hcol];
      const float ta = tauA[hcol];
      const float alpha = expf(-1.f / tm);
      const float rho   = expf(-1.f / ta);
      float cnt = 0.f;
      #pragma unroll
      for (int v = 0; v < 8; ++v) {
        const int b = mt * 16 + v + 8 * lh;
        const size_t idx = (size_t)b * HDIM + hcol;
        const float Ival = acc[v];
        const float uo = u_old[idx];
        const float ao = a_old[idx];
        const float zo = Acat[(size_t)b * KP + INDIM + hcol];
        const float an = rho * ao + (1.f - rho) * zo;
        const float un = alpha * uo + (1.f - alpha) * Ival - zo * (B0C + BETAC * ao);
        const float vv = un - (B0C + BETAC * an);
        const float zn = (vv > 0.f) ? 1.f : 0.f;
        u_new[idx] = un;
        a_new[idx] = an;
        Znew[idx]  = zn;
        if (periodic) { Iws[idx] = Ival; Zold[idx] = zo; }  // checkpoint for grads
        cnt += zn;
      }
      redS[tid] = cnt;
      __syncthreads();
      if (tid == 0) {            // exact: integer-valued, total < 2^24
        float s = 0.f;
        for (int i = 0; i < NTHR; ++i) s += redS[i];
        atomicAdd(nspk, s);
      }
    }
    grid_sync(bar);

    // ---- LI readout + output write; non-update steps also rebuild A_cat(t+1) ----
    if (gtid < BATCH * ODIM) {
      const int b = gtid / ODIM, o = gtid - b * ODIM;
      const float* zr = Znew + (size_t)b * HDIM;
      float s = 0.f;
      #pragma unroll 4
      for (int h = 0; h < HDIM; ++h) s += zr[h] * Wo[h * ODIM + o];
      const float to  = tauO[o];
      const float ao_ = expf(-1.f / to);
      const float oun = ao_ * ou_old[gtid] + (1.f - ao_) * s;
      rbuf[gtid]   = s;
      ou_new[gtid] = oun;
      out[(size_t)t * (BATCH * ODIM) + gtid] = oun;
    }
    if (!periodic) {
      const float* xt = x + (size_t)(t + 1) * (BATCH * INDIM);
      for (int i = gtid; i < BATCH * INDIM; i += GTHREADS) {
        int b = i / INDIM, j = i - b * INDIM;
        Acat[(size_t)b * KP + j] = xt[i];
      }
      for (int i = gtid; i < BATCH * HDIM; i += GTHREADS) {
        int b = i >> 9, h = i & (HDIM - 1);
        Acat[(size_t)b * KP + INDIM + h] = Znew[i];
      }
    }
    grid_sync(bar);

    // ---- loss: computed redundantly per WG (bitwise-identical fixed order) ----
    if (tid < BATCH) {
      const float* ob = ou_new + tid * ODIM;
      float m = ob[0];
      for (int o = 1; o < ODIM; ++o) m = fmaxf(m, ob[o]);
      float se = 0.f;
      for (int o = 0; o < ODIM; ++o) se += expf(ob[o] - m);
      const int tg = y[t * BATCH + tid];
      lossS[tid] = logf(se) + m - ob[tg];
    }
    __syncthreads();
    if (tid == 0) {
      float s = 0.f;
      for (int i = 0; i < BATCH; ++i) s += lossS[i];
      bcastS = s * (1.f / BATCH);
    }
    __syncthreads();
    const float sl = bcastS;
    const bool should = periodic && (lv > 1e-7f);   // uses loss_val BEFORE this step's add
    tot += sl;
    lv = should ? 0.f : (lv + sl);

    if (periodic) {
      if (should) {
        // U1: g_ou = (softmax(ou1) - onehot)/B ; cache pre-update decays
        if (gtid < BATCH * ODIM) {
          const int b = gtid / ODIM, o = gtid - b * ODIM;
          const float* ob = ou_new + b * ODIM;
          float m = ob[0];
          for (int oo = 1; oo < ODIM; ++oo) m = fmaxf(m, ob[oo]);
          float se = 0.f;
          for (int oo = 0; oo < ODIM; ++oo) se += expf(ob[oo] - m);
          const int tg = y[t * BATCH + b];
          const float p = expf(ob[o] - m) / se;
          gou[gtid] = (p - ((o == tg) ? 1.f : 0.f)) * (1.f / BATCH);
        }
        if (gtid < HDIM) {
          alC[gtid] = expf(-1.f / tauM[gtid]);
          rhC[gtid] = expf(-1.f / tauA[gtid]);
        }
        if (gtid < ODIM) aoC[gtid] = expf(-1.f / tauO[gtid]);
        grid_sync(bar);

        // U2: gv = (g_ou @ (W_o*(1-alpha_o))^T) * surrogate'(v)
        for (int i = gtid; i < BATCH * HDIM; i += GTHREADS) {
          const int b = i >> 9, h = i & (HDIM - 1);
          const float* gb = gou + b * ODIM;
          float gz = 0.f;
          for (int o = 0; o < ODIM; ++o)
            gz += gb[o] * Wo[h * ODIM + o] * (1.f - aoC[o]);
          const float vv = u_new[i] - (B0C + BETAC * a_new[i]);
          const float s  = 1.f / (1.f + expf(-SLOPEC * vv));
          gv[i] = gz * SLOPEC * s * (1.f - s);
        }
        grid_sync(bar);

        // U3: apply SGD updates in place
        for (int i = gtid; i < KC * HDIM; i += GTHREADS) {
          const int k = i / HDIM, h = i - k * HDIM;
          float dot = 0.f;
          if (k < INDIM) {
            for (int b = 0; b < BATCH; ++b)
              dot += Acat[(size_t)b * KP + k] * gv[b * HDIM + h];
          } else {
            const int hz = k - INDIM;
            for (int b = 0; b < BATCH; ++b)
              dot += Zold[b * HDIM + hz] * gv[b * HDIM + h];
          }
          WhT[(size_t)h * KP + k] -= LRC * (1.f - alC[h]) * dot;
        }
        for (int i = gtid; i < HDIM * ODIM; i += GTHREADS) {
          const int h = i / ODIM, o = i - h * ODIM;
          float dot = 0.f;
          for (int b = 0; b < BATCH; ++b)
            dot += Znew[b * HDIM + h] * gou[b * ODIM + o];
          Wo[i] -= LRC * (1.f - aoC[o]) * dot;
        }
        if (gtid < HDIM) {
          const int h = gtid;
          float s1 = 0.f, s2 = 0.f;
          for (int b = 0; b < BATCH; ++b) {
            const float g = gv[b * HDIM + h];
            s1 += g * (u_old[b * HDIM + h] - Iws[b * HDIM + h]);
            s2 += (-BETAC) * g * (a_old[b * HDIM + h] - Zold[b * HDIM + h]);
          }
          const float tm = tauM[h];
          tauM[h] = tm - LRC * (alC[h] / (tm * tm)) * s1;
          const float ta = tauA[h];
          tauA[h] = ta - LRC * (rhC[h] / (ta * ta)) * s2;
        }
        if (gtid < ODIM) {
          const int o = gtid;
          float s = 0.f;
          for (int b = 0; b < BATCH; ++b)
            s += gou[b * ODIM + o] * (ou_old[b * ODIM + o] - rbuf[b * ODIM + o]);
          const float to = tauO[o];
          tauO[o] = to - LRC * (aoC[o] / (to * to)) * s;
        }
        grid_sync(bar);
      }
      // deferred A_cat rebuild (A_cat/Zold were needed for the gradient)
      if (t + 1 < T_STEPS) {
        const float* xt = x + (size_t)(t + 1) * (BATCH * INDIM);
        for (int i = gtid; i < BATCH * INDIM; i += GTHREADS) {
          int b = i / INDIM, j = i - b * INDIM;
          Acat[(size_t)b * KP + j] = xt[i];
        }
        for (int i = gtid; i < BATCH * HDIM; i += GTHREADS) {
          int b = i >> 9, h = i & (HDIM - 1);
          Acat[(size_t)b * KP + INDIM + h] = Znew[i];
        }
      }
      grid_sync(bar);
    }
  }

  if (blockIdx.x == 0 && tid == 0) {
    out[(size_t)T_STEPS * BATCH * ODIM]     = tot;     // total loss
    out[(size_t)T_STEPS * BATCH * ODIM + 1] = *nspk;   // spike count
  }
}

extern "C" void kernel_launch(void* const* d_in, const int* in_sizes, int n_in,
                              void* d_out, int out_size, void* d_ws, size_t ws_size,
                              hipStream_t stream) {
  (void)in_sizes; (void)n_in; (void)out_size; (void)ws_size;
  const float* x   = (const float*)d_in[0];
  const int*   y   = (const int*)  d_in[1];
  const float* Wh  = (const float*)d_in[2];
  const float* tm  = (const float*)d_in[3];
  const float* ta  = (const float*)d_in[4];
  const float* Wo  = (const float*)d_in[5];
  const float* tmo = (const float*)d_in[6];
  float* ws  = (float*)d_ws;
  float* out = (float*)d_out;

  alif_init<<<dim3(512), dim3(256), 0, stream>>>(Wh, tm, ta, Wo, tmo, ws);
  alif_run<<<dim3(NWG), dim3(NTHR), 0, stream>>>(x, y, out, ws);
}